// toRGB_68487548502382
// MI455X (gfx1250) — compile-verified
//
#include <hip/hip_runtime.h>

// ---------------------------------------------------------------------------
// StyleGAN2 toRGB: style-FC + modulated 1x1 conv + bias + LeakyReLU.
// Memory-bound (256 MB x read once @ 23.3 TB/s dominates); main kernel
// streams x with NT loads and runs the contraction on v_wmma_f32_16x16x4_f32
// (fp32 matrix pipe, K=128 as 32 chunks of 4, N=3 zero-padded to 16).
// ---------------------------------------------------------------------------

typedef __attribute__((ext_vector_type(2))) float v2f;
typedef __attribute__((ext_vector_type(8))) float v8f;

#define CIN        128
#define COUT       3
#define WDIM       512
#define NB         8
#define PIXPB      (256 * 256)
#define LRELU      0.2f
#define FC_SCALE   0.04419417382415922f   // 1/sqrt(512)
#define CONV_SCALE 0.08838834764831845f   // 1/sqrt(128)
#define ACT_GAIN   1.4142135623730951f    // sqrt(2)

// ---------------------------------------------------------------------------
// Prologue: style[b][i] = leaky(w[b]·fc_w[:,i]*FC_SCALE + fc_b[i]) * sqrt(2)
// 8 blocks x 128 threads; 0.5 MFLOP total — negligible.
// ---------------------------------------------------------------------------
__global__ void __launch_bounds__(CIN)
style_kernel(const float* __restrict__ w, const float* __restrict__ fc_w,
             const float* __restrict__ fc_b, float* __restrict__ style) {
    const int b = blockIdx.x;
    const int i = threadIdx.x;                 // output channel 0..127
    const float* wb = w + b * WDIM;
    float acc = 0.f;
    for (int k = 0; k < WDIM; ++k)
        acc = fmaf(wb[k], fc_w[k * CIN + i], acc);
    float s = acc * FC_SCALE + fc_b[i];
    s = (s > 0.f) ? s : LRELU * s;
    style[b * CIN + i] = s * ACT_GAIN;
}

// ---------------------------------------------------------------------------
// Main kernel: grid (64, 8); block = 256 threads = 8 waves.
// Each wave computes 8 tiles of 16 pixels via 32x v_wmma_f32_16x16x4_f32.
// ---------------------------------------------------------------------------
__global__ void __launch_bounds__(256)
torgb_kernel(const float* __restrict__ x, const float* __restrict__ style,
             const float* __restrict__ conv_w, const float* __restrict__ bias,
             float* __restrict__ out) {
    // B-fragment table: sW[j][n][p] = W_eff[2j+p][n], zero-padded for n>=3.
    // Pairs (p=0,1) adjacent -> each B fragment is one ds_load_b64.
    __shared__ float sW[64 * 16 * 2];          // 8 KB

    const int b   = blockIdx.y;
    const int tid = threadIdx.x;

    for (int idx = tid; idx < 64 * 32; idx += 256) {
        const int j = idx >> 5;                // chunk-half: rows 2j, 2j+1
        const int rem = idx & 31;
        const int n = rem >> 1;
        const int p = rem & 1;
        const int i = 2 * j + p;               // input channel
        float v = 0.f;
        if (n < COUT) v = style[b * CIN + i] * conv_w[i * COUT + n] * CONV_SCALE;
        sW[idx] = v;
    }
    __syncthreads();

    const int wave   = tid >> 5;
    const int lane   = tid & 31;
    const int laneHi = lane >> 4;              // 0 or 1 (lane half)
    const int n      = lane & 15;              // N column / pixel-in-tile row

    // Load all 32 B fragments once per wave; held in VGPRs across tiles.
    // ISA layout (B 4x16 f32): V0 = rows {4k+2*laneHi}, V1 = rows {+1}.
    v2f bfrag[32];
#pragma unroll
    for (int k = 0; k < 32; ++k) {
        const int j = 2 * k + laneHi;
        bfrag[k] = *(const v2f*)&sW[(j * 16 + n) * 2];
    }

    const float bn = (n < COUT) ? bias[n] : 0.f;

    const float* xb = x   + (size_t)b * PIXPB * CIN;
    float*       ob = out + (size_t)b * PIXPB * COUT;

    const int tile0 = (blockIdx.x * 8 + wave) * 8;  // 8 tiles per wave

#pragma unroll 1
    for (int t = 0; t < 8; ++t) {
        const int p0 = (tile0 + t) * 16;       // first pixel of this 16-row tile

        // A fragments: lane holds pixel p0+n, channels (4k+2*laneHi, +1).
        // x is streamed exactly once -> non-temporal loads (x > L2 capacity).
        const float* xp = xb + (size_t)(p0 + n) * CIN + 2 * laneHi;
        v2f afrag[32];
#pragma unroll
        for (int k = 0; k < 32; ++k)
            afrag[k] = __builtin_nontemporal_load((const v2f*)(xp + 4 * k));

        v8f c = {};                            // fp32 accumulator, 16x16
#pragma unroll
        for (int k = 0; k < 32; ++k)
            c = __builtin_amdgcn_wmma_f32_16x16x4_f32(
                    /*neg_a=*/false, afrag[k], /*neg_b=*/false, bfrag[k],
                    /*c_mod=*/(short)0, c, /*reuse_a=*/false, /*reuse_b=*/false);

        // D layout: c[v] holds (M = v + 8*laneHi, N = n). Only N<3 is real.
        if (n < COUT) {
#pragma unroll
            for (int v = 0; v < 8; ++v) {
                const int pix = p0 + v + 8 * laneHi;
                float y = c[v] + bn;
                y = (y > 0.f) ? y : LRELU * y;
                __builtin_nontemporal_store(y, &ob[(size_t)pix * COUT + n]);
            }
        }
    }
}

extern "C" void kernel_launch(void* const* d_in, const int* in_sizes, int n_in,
                              void* d_out, int out_size, void* d_ws, size_t ws_size,
                              hipStream_t stream) {
    const float* x      = (const float*)d_in[0];
    const float* w      = (const float*)d_in[1];
    const float* fc_w   = (const float*)d_in[2];
    const float* fc_b   = (const float*)d_in[3];
    const float* conv_w = (const float*)d_in[4];
    const float* bias   = (const float*)d_in[5];
    float*       out    = (float*)d_out;
    float*       style  = (float*)d_ws;        // 8*128 floats = 4 KB scratch

    style_kernel<<<NB, CIN, 0, stream>>>(w, fc_w, fc_b, style);

    dim3 grid(64, NB);                         // 64 blocks/batch * 1024 px/block
    torgb_kernel<<<grid, 256, 0, stream>>>(x, style, conv_w, bias, out);
}